// EquivariantLayerNorm_29154238005848
// MI455X (gfx1250) — compile-verified
//
#include <hip/hip_runtime.h>
#include <stdint.h>

#define WAVES  8
#define ROW    480
#define ROWV4  120   // float4 per row (1920 bytes)

typedef float v4f __attribute__((ext_vector_type(4)));

__device__ __forceinline__ void nt_store4(float* p, float a, float b, float c, float d) {
    v4f t = {a, b, c, d};
    __builtin_nontemporal_store(t, (v4f*)p);
}

struct SMem {
    float4 buf[WAVES][2][ROWV4];  // per-wave double-buffered row staging
    float4 w4[ROWV4];             // repeat-expanded weight (480 floats)
    float4 b4[32];                // bias (128 floats)
};

__global__ __launch_bounds__(256) void eqln_kernel(
    const float* __restrict__ x, const float* __restrict__ weight,
    const float* __restrict__ bias, float* __restrict__ out, int nrows)
{
    __shared__ SMem sm;
    const int tid = threadIdx.x;

    // Build repeat-expanded weight table once per block:
    // cols 0..127 -> w[c]; 128..319 -> w[128+(c-128)/3]; 320..479 -> w[192+(c-320)/5].
    float* sw = (float*)sm.w4;
    for (int i = tid; i < ROW; i += 256) {
        int f = (i < 128) ? i : ((i < 320) ? (128 + (i - 128) / 3) : (192 + (i - 320) / 5));
        sw[i] = weight[f];
    }
    float* sb = (float*)sm.b4;
    if (tid < 128) sb[tid] = bias[tid];
    __syncthreads();

    const int wid  = tid >> 5;
    const int lane = tid & 31;
    const int stride = gridDim.x * WAVES;
    int row = blockIdx.x * WAVES + wid;

    // Low 32 bits of the flat address of an LDS object == wave-relative LDS byte offset.
    const uint32_t lds0 = (uint32_t)(uintptr_t)(&sm.buf[wid][0][0]);
    const uint32_t lds1 = (uint32_t)(uintptr_t)(&sm.buf[wid][1][0]);

    // One row (1920B) = 4 async b128 loads; INST_OFFSET advances both the global
    // and the LDS address, so one vaddr pair + one LDS-offset VGPR cover the row.
    // 4th load runs under partial EXEC (lanes 0..23): 1920 = 3*512 + 24*16.
    auto issue_row = [&](int r, uint32_t ldsb) {
        uint64_t ga = (uint64_t)(uintptr_t)(x + (size_t)r * ROW) + (uint64_t)(lane * 16);
        uint32_t la = ldsb + (uint32_t)(lane * 16);
        asm volatile("global_load_async_to_lds_b128 %0, %1, off th:TH_LOAD_NT"
                     :: "v"(la), "v"(ga) : "memory");
        asm volatile("global_load_async_to_lds_b128 %0, %1, off offset:512 th:TH_LOAD_NT"
                     :: "v"(la), "v"(ga) : "memory");
        asm volatile("global_load_async_to_lds_b128 %0, %1, off offset:1024 th:TH_LOAD_NT"
                     :: "v"(la), "v"(ga) : "memory");
        if (lane < 24)
            asm volatile("global_load_async_to_lds_b128 %0, %1, off offset:1536 th:TH_LOAD_NT"
                         :: "v"(la), "v"(ga) : "memory");
    };

    if (row < nrows) issue_row(row, lds0);
    int cur = 0;

    for (; row < nrows; row += stride) {
        const int nxt = row + stride;
        if (nxt < nrows) {
            issue_row(nxt, cur ? lds0 : lds1);             // prefetch into other buffer
            asm volatile("s_wait_asynccnt 4" ::: "memory"); // oldest 4 (this row) done
        } else {
            asm volatile("s_wait_asynccnt 0" ::: "memory");
        }

        const float4* b = &sm.buf[wid][cur][0];
        float4 v0 = b[lane];
        float4 v1 = b[lane + 32];
        float4 v2 = b[lane + 64];
        float4 v3 = make_float4(0.f, 0.f, 0.f, 0.f);
        if (lane < 24) v3 = b[lane + 96];

        // Single fused wave reduction:
        //   s1 = sum of scalar block (cols 0..127 == the j=0 float4 of every lane)
        //   s2 = sum of ALL 480 raw squares
        // Then  sum((x-m)^2 over scalars) + sum(x^2 over rest) = s2 - s1^2/128,
        // and   mean(norm_sq over 224 channels) = (s2 - s1^2/128) / 224.
        float s1 = v0.x + v0.y + v0.z + v0.w;
        float s2 = v0.x*v0.x + v0.y*v0.y + v0.z*v0.z + v0.w*v0.w
                 + v1.x*v1.x + v1.y*v1.y + v1.z*v1.z + v1.w*v1.w
                 + v2.x*v2.x + v2.y*v2.y + v2.z*v2.z + v2.w*v2.w
                 + v3.x*v3.x + v3.y*v3.y + v3.z*v3.z + v3.w*v3.w;
#pragma unroll
        for (int off = 16; off > 0; off >>= 1) {
            s1 += __shfl_xor(s1, off, 32);
            s2 += __shfl_xor(s2, off, 32);
        }
        const float m   = s1 * (1.0f / 128.0f);
        const float ss  = s2 - s1 * s1 * (1.0f / 128.0f);
        const float inv = rsqrtf(ss * (1.0f / 224.0f));

        v0.x -= m; v0.y -= m; v0.z -= m; v0.w -= m;

        float* orow = out + (size_t)row * ROW;
        const float4 w0 = sm.w4[lane];
        const float4 w1 = sm.w4[lane + 32];
        const float4 w2 = sm.w4[lane + 64];
        const float4 bb = sm.b4[lane];

        nt_store4(orow + 4 * lane,
                  v0.x * inv * w0.x + bb.x, v0.y * inv * w0.y + bb.y,
                  v0.z * inv * w0.z + bb.z, v0.w * inv * w0.w + bb.w);
        nt_store4(orow + 4 * (lane + 32),
                  v1.x * inv * w1.x, v1.y * inv * w1.y,
                  v1.z * inv * w1.z, v1.w * inv * w1.w);
        nt_store4(orow + 4 * (lane + 64),
                  v2.x * inv * w2.x, v2.y * inv * w2.y,
                  v2.z * inv * w2.z, v2.w * inv * w2.w);
        if (lane < 24) {
            const float4 w3 = sm.w4[lane + 96];
            nt_store4(orow + 4 * (lane + 96),
                      v3.x * inv * w3.x, v3.y * inv * w3.y,
                      v3.z * inv * w3.z, v3.w * inv * w3.w);
        }
        cur ^= 1;
    }
}

extern "C" void kernel_launch(void* const* d_in, const int* in_sizes, int n_in,
                              void* d_out, int out_size, void* d_ws, size_t ws_size,
                              hipStream_t stream) {
    const float* x      = (const float*)d_in[0];
    const float* weight = (const float*)d_in[1];
    const float* bias   = (const float*)d_in[2];
    float* out          = (float*)d_out;

    const int nrows = in_sizes[0] / ROW;      // 131072
    int grid = (nrows + WAVES - 1) / WAVES;
    if (grid > 4096) grid = 4096;             // ~4 rows per wave -> pipelined double buffer
    eqln_kernel<<<grid, 256, 0, stream>>>(x, weight, bias, out, nrows);
}